// IterNorm_11527692222570
// MI455X (gfx1250) — compile-verified
//
#include <hip/hip_runtime.h>
#include <hip/hip_bf16.h>

// ---------------------------------------------------------------------------
// IterNorm (iterative whitening) for X(32,256,56,56) fp32, G=4 groups, D=64.
// Memory-bound: ~0.31 GB HBM traffic -> ~13us floor @ 23.3 TB/s.
// Matrix work: V_WMMA_F32_16X16X4_F32 (fp32 WMMA, wave32).
// LDS staging: Tensor Data Mover (tensor_load_to_lds + s_wait_tensorcnt).
// ---------------------------------------------------------------------------

typedef float v2f __attribute__((ext_vector_type(2)));
typedef float v8f __attribute__((ext_vector_type(8)));
typedef unsigned int v4u __attribute__((ext_vector_type(4)));
typedef int v4i __attribute__((ext_vector_type(4)));
typedef int v8i __attribute__((ext_vector_type(8)));

#define NB    32          // batch
#define CCH   256         // channels
#define HW    3136        // 56*56
#define GRP   4
#define DCH   64
#define MTOT  100352.0f   // NB*HW
#define EPSV  1e-5f
#define NS_T  5

#define K1_CHUNK 196      // 3136 = 16 * 196, 196 % 4 == 0
#define K1_NCHUNK 16

__device__ __forceinline__ v8f wmma_f32_16x16x4(v2f a, v2f b, v8f c) {
    // D = A(16x4,f32) * B(4x16,f32) + C(16x16,f32)
    return __builtin_amdgcn_wmma_f32_16x16x4_f32(
        /*neg_a=*/false, a, /*neg_b=*/false, b,
        /*c_mod=*/(short)0, c, /*reuse_a=*/false, /*reuse_b=*/false);
}

// LDS byte offset of a __shared__ object (generic -> addrspace(3) -> int).
__device__ __forceinline__ unsigned lds_addr_of(const void* p) {
    return (unsigned)(unsigned long long)
           (__attribute__((address_space(3))) const void*)p;
}

// ---------------------------------------------------------------------------
// TDM: stage a 2D tile (tile_h rows x tile_w fp32 elems, row stride
// row_stride_elems) from global memory into LDS (linear row-major).
// D# layout per CDNA5 ISA 8.3-8.4. Issue from ONE wave; TENSORcnt is per-wave.
// ---------------------------------------------------------------------------
__device__ __forceinline__ void tdm_load_2d(unsigned lds_byte_off,
                                            const float* gsrc,
                                            unsigned tile_w, unsigned tile_h,
                                            unsigned row_stride_elems) {
    unsigned long long ga = (unsigned long long)gsrc;
    v4u g0;
    g0.x = 1u;                                     // count=1 (valid), load, user
    g0.y = lds_byte_off;                           // lds_addr (bytes)
    g0.z = (unsigned)(ga & 0xFFFFFFFFull);         // global_addr[31:0]
    g0.w = (unsigned)((ga >> 32) & 0x1FFFFFFu)     // global_addr[56:32]
         | (2u << 30);                             // type = 2 ("image")
    v8i g1;
    g1[0] = (int)(2u << 16);                       // wg_mask=0, data_size=2 (4B)
    g1[1] = (int)((tile_w & 0xFFFFu) << 16);       // tensor_dim0[15:0]
    g1[2] = (int)(((tile_w >> 16) & 0xFFFFu)       // tensor_dim0[31:16]
         | ((tile_h & 0xFFFFu) << 16));            // tensor_dim1[15:0]
    g1[3] = (int)(((tile_h >> 16) & 0xFFFFu)       // tensor_dim1[31:16]
         | ((tile_w & 0xFFFFu) << 16));            // tile_dim0
    g1[4] = (int)(tile_h & 0xFFFFu);               // tile_dim1 (tile_dim2 = 0)
    g1[5] = (int)row_stride_elems;                 // tensor_dim0_stride[31:0]
    g1[6] = 0;                                     // d0_stride hi / d1_stride lo
    g1[7] = 0;                                     // d1_stride hi
    v4i gz = {0, 0, 0, 0};
#if defined(__clang_major__) && (__clang_major__ >= 23)
    v8i gz8 = {0, 0, 0, 0, 0, 0, 0, 0};
    __builtin_amdgcn_tensor_load_to_lds(g0, g1, gz, gz, gz8, 0);
#else
    __builtin_amdgcn_tensor_load_to_lds(g0, g1, gz, gz, 0);
#endif
}

// ---------------------------------------------------------------------------
// Kernel 0: zero the accumulators (Gram + channel sums) in workspace.
// ---------------------------------------------------------------------------
__global__ void k_zero(float* __restrict__ p, int n) {
    int i = blockIdx.x * blockDim.x + threadIdx.x;
    if (i < n) p[i] = 0.0f;
}

// ---------------------------------------------------------------------------
// Kernel 1: per-group raw Gram  G[g] += x_slab @ x_slab^T  and channel sums.
// One WG = (group g, image n, 196-col chunk). TDM stages the 64x196 slab;
// 16 waves each own a 16x16 Gram tile: 49 x WMMA(K=4).
// LDS stride 196 == 4 (mod 64 banks) -> 16-lane column reads conflict-free.
// ---------------------------------------------------------------------------
__global__ void __launch_bounds__(512)
k_gram(const float* __restrict__ X,
       float* __restrict__ gram,    // [GRP][64][64]
       float* __restrict__ sums) {  // [GRP][64]
    __shared__ __align__(16) float xs[DCH * K1_CHUNK];

    int blk   = blockIdx.x;               // g*32*16 + n*16 + chunk
    int chunk = blk & (K1_NCHUNK - 1);
    int n     = (blk >> 4) & 31;
    int g     = blk >> 9;
    int tid   = threadIdx.x;

    long base = ((long)n * CCH + (long)g * DCH) * HW + (long)chunk * K1_CHUNK;

    if (tid < 32) {   // wave 0 issues the DMA and waits on its TENSORcnt
        tdm_load_2d(lds_addr_of(xs), X + base, K1_CHUNK, DCH, HW);
        __builtin_amdgcn_s_wait_tensorcnt(0);
    }
    __syncthreads();

    // Per-channel partial sums (fused mean: Sigma = E[xx^T] - mu mu^T).
    if (tid < DCH) {
        float s = 0.0f;
        #pragma unroll 4
        for (int k = 0; k < K1_CHUNK; ++k) s += xs[tid * K1_CHUNK + k];
        atomicAdd(&sums[g * DCH + tid], s);
    }

    // WMMA Gram tiles.
    int wave  = tid >> 5;
    int lane  = tid & 31;
    int r0    = (wave >> 2) * 16;
    int c0    = (wave & 3) * 16;
    int l15   = lane & 15;
    int khalf = (lane >> 4) * 2;          // upper half-wave holds K+2, K+3

    const float* ar = xs + (r0 + l15) * K1_CHUNK + khalf;
    const float* br = xs + (c0 + l15) * K1_CHUNK + khalf;

    v8f acc = {};
    for (int k0 = 0; k0 < K1_CHUNK; k0 += 4) {
        v2f a, b;
        a.x = ar[k0];      // A[m][k]
        a.y = ar[k0 + 1];
        b.x = br[k0];      // B[k][e] = x[e][k]  (Gram)
        b.y = br[k0 + 1];
        acc = wmma_f32_16x16x4(a, b, acc);
    }

    // C/D layout: VGPR j -> row r0+j (lanes 0-15), r0+j+8 (lanes 16-31).
    int rowadd = (lane >> 4) << 3;
    #pragma unroll
    for (int j = 0; j < 8; ++j) {
        int row = r0 + j + rowadd;
        int col = c0 + l15;
        atomicAdd(&gram[g * 4096 + row * 64 + col], acc[j]);
    }
}

// ---------------------------------------------------------------------------
// Kernel 2: finalize Sigma, trace-normalize, Newton-Schulz (T=5),
// emit wm and wmmean = wm @ mean (for the fused epilogue in k_apply).
// ---------------------------------------------------------------------------
__device__ __forceinline__ void mm64(float* __restrict__ D,
                                     const float* __restrict__ A,
                                     const float* __restrict__ B, int tid) {
    __syncthreads();
    for (int idx = tid; idx < 4096; idx += 256) {
        int r = idx >> 6, c = idx & 63;
        const float* arow = A + r * 64;
        float s = 0.0f;
        #pragma unroll 8
        for (int k = 0; k < 64; ++k) s = fmaf(arow[k], B[k * 64 + c], s);
        D[idx] = s;
    }
    __syncthreads();
}

__global__ void __launch_bounds__(256)
k_newton(const float* __restrict__ gram,
         const float* __restrict__ sums,
         float* __restrict__ wmmeanOut,  // [GRP][64]
         float* __restrict__ wmOut) {    // [GRP][64][64]
    __shared__ float Sn[4096];
    __shared__ float P [4096];
    __shared__ float T1[4096];
    __shared__ float T2[4096];
    __shared__ float meanS[DCH];
    __shared__ float rTrS;

    int g   = blockIdx.x;
    int tid = threadIdx.x;
    const float invM = 1.0f / MTOT;

    if (tid < DCH) meanS[tid] = sums[g * DCH + tid] * invM;
    __syncthreads();

    for (int idx = tid; idx < 4096; idx += 256) {
        int r = idx >> 6, c = idx & 63;
        float s = gram[g * 4096 + idx] * invM - meanS[r] * meanS[c];
        if (r == c) s += EPSV;
        Sn[idx] = s;
        P[idx]  = (r == c) ? 1.0f : 0.0f;
    }
    __syncthreads();

    if (tid == 0) {
        float tr = 0.0f;
        #pragma unroll
        for (int d = 0; d < DCH; ++d) tr += Sn[d * 64 + d];
        rTrS = 1.0f / tr;
    }
    __syncthreads();
    float rTr = rTrS;
    for (int idx = tid; idx < 4096; idx += 256) Sn[idx] *= rTr;

    for (int it = 0; it < NS_T; ++it) {
        mm64(T1, P,  P,  tid);     // P^2
        mm64(T2, T1, P,  tid);     // P^3
        mm64(T1, T2, Sn, tid);     // P^3 * Sn
        for (int idx = tid; idx < 4096; idx += 256)
            P[idx] = 1.5f * P[idx] - 0.5f * T1[idx];
        __syncthreads();
    }

    float sr = sqrtf(rTr);
    for (int idx = tid; idx < 4096; idx += 256)
        wmOut[g * 4096 + idx] = P[idx] * sr;

    if (tid < DCH) {               // wmmean[d] = sum_e wm[d][e] * mean[e]
        float s = 0.0f;
        #pragma unroll 8
        for (int e = 0; e < DCH; ++e)
            s = fmaf(P[tid * 64 + e] * sr, meanS[e], s);
        wmmeanOut[g * DCH + tid] = s;
    }
}

// ---------------------------------------------------------------------------
// Kernel 3: out = (wm @ x - wm @ mean) * weight + bias.
// TDM stages the raw 64x64 x-block (mean folded into epilogue); wm staged
// 65-padded (its A-frag reads are column-pattern). 16 waves x 16 WMMA(K=4).
// ---------------------------------------------------------------------------
__global__ void __launch_bounds__(512)
k_apply(const float* __restrict__ X,
        const float* __restrict__ wm,      // [GRP][64][64]
        const float* __restrict__ wmmean,  // [GRP][64]
        const float* __restrict__ weight,  // [256]
        const float* __restrict__ bias,    // [256]
        float* __restrict__ out) {
    __shared__ __align__(16) float xcS[DCH * DCH];   // raw x block (K x N)
    __shared__ float wmS[DCH][DCH + 1];

    int blk = blockIdx.x;                  // (g*32 + n)*49 + cb
    int cb  = blk % 49;
    int t   = blk / 49;
    int n   = t & 31;
    int g   = t >> 5;
    int tid = threadIdx.x;

    int  hw0  = cb * 64;
    long base = ((long)n * CCH + (long)g * DCH) * HW + hw0;

    if (tid < 32) {   // wave 0: DMA the x block while everyone stages wm
        tdm_load_2d(lds_addr_of(xcS), X + base, DCH, DCH, HW);
    }
    for (int idx = tid; idx < 4096; idx += 512) {
        int r = idx >> 6, c = idx & 63;
        wmS[r][c] = wm[g * 4096 + idx];
    }
    if (tid < 32) __builtin_amdgcn_s_wait_tensorcnt(0);
    __syncthreads();

    int wave  = tid >> 5;
    int lane  = tid & 31;
    int r0    = (wave >> 2) * 16;
    int c0    = (wave & 3) * 16;
    int l15   = lane & 15;
    int khalf = (lane >> 4) * 2;

    v8f acc = {};
    #pragma unroll
    for (int k0 = 0; k0 < DCH; k0 += 4) {
        v2f a, b;
        a.x = wmS[r0 + l15][k0 + khalf];              // A = wm rows
        a.y = wmS[r0 + l15][k0 + khalf + 1];
        b.x = xcS[(k0 + khalf)     * DCH + c0 + l15]; // B = x (K x N)
        b.y = xcS[(k0 + khalf + 1) * DCH + c0 + l15];
        acc = wmma_f32_16x16x4(a, b, acc);
    }

    int rowadd = (lane >> 4) << 3;
    #pragma unroll
    for (int j = 0; j < 8; ++j) {
        int row = r0 + j + rowadd;
        int ch  = g * DCH + row;
        float v = (acc[j] - wmmean[ch]) * weight[ch] + bias[ch];
        out[((long)n * CCH + ch) * HW + hw0 + c0 + l15] = v;
    }
}

// ---------------------------------------------------------------------------
// Launcher
// ---------------------------------------------------------------------------
extern "C" void kernel_launch(void* const* d_in, const int* in_sizes, int n_in,
                              void* d_out, int out_size, void* d_ws, size_t ws_size,
                              hipStream_t stream) {
    const float* X      = (const float*)d_in[0];
    const float* weight = (const float*)d_in[1];
    const float* bias   = (const float*)d_in[2];
    float* out = (float*)d_out;

    float* ws     = (float*)d_ws;
    float* gram   = ws;                 // 4*4096 = 16384 floats
    float* sums   = ws + 16384;         // 256
    float* wmmean = ws + 16640;         // 256
    float* wmbuf  = ws + 16896;         // 4*4096 = 16384
    const int nzero = 16640;            // gram + sums must start at 0

    k_zero  <<<(nzero + 255) / 256, 256, 0, stream>>>(ws, nzero);
    k_gram  <<<GRP * NB * K1_NCHUNK, 512, 0, stream>>>(X, gram, sums);
    k_newton<<<GRP, 256, 0, stream>>>(gram, sums, wmmean, wmbuf);
    k_apply <<<GRP * NB * 49, 512, 0, stream>>>(X, wmbuf, wmmean, weight, bias, out);
}